// EdgeDecoder_76347338653860
// MI455X (gfx1250) — compile-verified
//
#include <hip/hip_runtime.h>
#include <hip/hip_bf16.h>
#include <stdint.h>

typedef float v2f __attribute__((ext_vector_type(2)));
typedef float v8f __attribute__((ext_vector_type(8)));

#define CDIM 64
#define ROWPAD 68                 // padded LDS row stride (floats): bank = (4r+c)%64, conflict-free
#define TILEF (16 * ROWPAD)       // floats per padded 16x64 tile buffer (4352 B)

// ---------------------------------------------------------------------------
// Async global->LDS copy (CDNA5 GLOBAL_LOAD_ASYNC_TO_LDS_B128, ASYNCcnt).
// ---------------------------------------------------------------------------
__device__ __forceinline__ void async_load_b128(uint32_t lds_off, const float* gaddr) {
  asm volatile("global_load_async_to_lds_b128 %0, %1, off"
               :: "v"(lds_off), "v"(gaddr)
               : "memory");
}

// Copy one 16x64 f32 tile (4 KB) into a padded LDS buffer: 8 coalesced 512B
// transfers (each lane moves one 16B chunk per issue).
__device__ __forceinline__ void load_tile_async(const float* gbase,
                                                const float* lbase, int lane) {
#pragma unroll
  for (int i = 0; i < 8; ++i) {
    const int g = i * 128 + lane * 4;        // flat float index in tile
    const int r = g >> 6, c = g & 63;        // row / channel
    async_load_b128((uint32_t)(uintptr_t)(lbase + r * ROWPAD + c), gbase + g);
  }
}

// ---------------------------------------------------------------------------
// Phase 1: per-node projections  sk[n] = h[n].Wk , si[n] = h[n].Wi
// D = A(16x4 weights) x B(4x16 node-columns) over 16 K-steps of
// V_WMMA_F32_16X16X4_F32 (full f32 precision). A row0 = Wk, row1 = Wi.
// ---------------------------------------------------------------------------
__global__ void __launch_bounds__(256) node_proj_wmma(
    const float* __restrict__ h, const float* __restrict__ W,
    float* __restrict__ sk, float* __restrict__ si, int n_tiles) {
  const int lane = threadIdx.x & 31;
  const int row  = lane & 15;
  const int half = lane >> 4;
  const int wave   = (blockIdx.x * blockDim.x + threadIdx.x) >> 5;
  const int nwaves = (gridDim.x * blockDim.x) >> 5;

  const float* wrow = (row == 1) ? (W + CDIM) : W;  // row1 = Wi, else Wk
  v2f a[16];
#pragma unroll
  for (int k = 0; k < 16; ++k) {
    a[k].x = wrow[4 * k + 2 * half];
    a[k].y = wrow[4 * k + 2 * half + 1];
  }

  for (int tile = wave; tile < n_tiles; tile += nwaves) {
    const int item = tile * 16 + row;
    const float* brow = h + (size_t)item * CDIM + 2 * half;
    v8f acc = {};
#pragma unroll
    for (int k = 0; k < 16; ++k) {
      v2f b;
      b.x = brow[4 * k];
      b.y = brow[4 * k + 1];
      acc = __builtin_amdgcn_wmma_f32_16x16x4_f32(
          false, a[k], false, b, (short)0, acc, false, false);
    }
    if (half == 0) {
      sk[item] = acc[0];   // D row M=0
      si[item] = acc[1];   // D row M=1
    }
  }
}

// ---------------------------------------------------------------------------
// Phase 2: per-edge output  p = e.We + sk[src] + si[dst] + bias
// e-stream staged via double-buffered async global->LDS b128 copies,
// WMMA B-fragments read from bank-conflict-free padded LDS (ds_load_b64).
// ---------------------------------------------------------------------------
__global__ void __launch_bounds__(128) edge_out_wmma(
    const float* __restrict__ e, const float* __restrict__ W,
    const float* __restrict__ bptr,
    const int* __restrict__ src, const int* __restrict__ dst,
    const float* __restrict__ sk, const float* __restrict__ si,
    float* __restrict__ p, int n_tiles) {
  __shared__ float smem[4 * 2 * TILEF];   // 4 waves x 2 buffers = 34,816 B
  const int lane  = threadIdx.x & 31;
  const int row   = lane & 15;
  const int half  = lane >> 4;
  const int wslot = threadIdx.x >> 5;
  float* const buf0 = smem + wslot * (2 * TILEF);
  float* const buf1 = buf0 + TILEF;

  const int wave   = (blockIdx.x * blockDim.x + threadIdx.x) >> 5;
  const int nwaves = (gridDim.x * blockDim.x) >> 5;

  const float* wrow = W + 2 * CDIM;  // We on every A-row
  v2f a[16];
#pragma unroll
  for (int k = 0; k < 16; ++k) {
    a[k].x = wrow[4 * k + 2 * half];
    a[k].y = wrow[4 * k + 2 * half + 1];
  }
  const float bias = bptr[0];

  int tile = wave;
  if (tile >= n_tiles) return;                 // wave-uniform
  load_tile_async(e + (size_t)tile * (16 * CDIM), buf0, lane);

  int cur = 0;
  while (tile < n_tiles) {                     // wave-uniform loop
    const int nxt = tile + nwaves;
    const float* cbuf = cur ? buf1 : buf0;
    if (nxt < n_tiles) {
      load_tile_async(e + (size_t)nxt * (16 * CDIM), cur ? buf0 : buf1, lane);
      asm volatile("s_wait_asynccnt 0x8" ::: "memory");  // older tile resident
    } else {
      asm volatile("s_wait_asynccnt 0x0" ::: "memory");
    }

    const float* brow = cbuf + row * ROWPAD + 2 * half;
    v8f acc = {};
#pragma unroll
    for (int k = 0; k < 16; ++k) {
      v2f b;
      b.x = brow[4 * k];      // ds_load_b64, all 64 banks hit exactly once
      b.y = brow[4 * k + 1];
      acc = __builtin_amdgcn_wmma_f32_16x16x4_f32(
          false, a[k], false, b, (short)0, acc, false, false);
    }
    if (half == 0) {
      const int item = tile * 16 + row;
      p[item] = acc[0] + sk[src[item]] + si[dst[item]] + bias;  // 4B L2 gathers
    }
    tile = nxt;
    cur ^= 1;
  }
}

// ---------------------------------------------------------------------------
// Scalar tails (only launched if counts are not multiples of 16).
// ---------------------------------------------------------------------------
__global__ void node_proj_tail(const float* __restrict__ h,
                               const float* __restrict__ W,
                               float* __restrict__ sk, float* __restrict__ si,
                               int start, int n_nodes) {
  int n = start + blockIdx.x * blockDim.x + threadIdx.x;
  if (n >= n_nodes) return;
  float vk = 0.f, vi = 0.f;
  for (int c = 0; c < CDIM; ++c) {
    float x = h[(size_t)n * CDIM + c];
    vk += x * W[c];
    vi += x * W[CDIM + c];
  }
  sk[n] = vk;
  si[n] = vi;
}

__global__ void edge_out_tail(const float* __restrict__ e,
                              const float* __restrict__ W,
                              const float* __restrict__ bptr,
                              const int* __restrict__ src,
                              const int* __restrict__ dst,
                              const float* __restrict__ sk,
                              const float* __restrict__ si,
                              float* __restrict__ p, int start, int n_edges) {
  int n = start + blockIdx.x * blockDim.x + threadIdx.x;
  if (n >= n_edges) return;
  float v = 0.f;
  for (int c = 0; c < CDIM; ++c) v += e[(size_t)n * CDIM + c] * W[2 * CDIM + c];
  p[n] = v + sk[src[n]] + si[dst[n]] + bptr[0];
}

extern "C" void kernel_launch(void* const* d_in, const int* in_sizes, int n_in,
                              void* d_out, int out_size, void* d_ws, size_t ws_size,
                              hipStream_t stream) {
  const float* h   = (const float*)d_in[0];
  const float* e   = (const float*)d_in[1];
  const float* W   = (const float*)d_in[2];
  const float* b   = (const float*)d_in[3];
  const int*   src = (const int*)d_in[4];
  const int*   dst = (const int*)d_in[5];
  float*       p   = (float*)d_out;

  const int n_nodes = in_sizes[0] / CDIM;
  const int n_edges = in_sizes[4];

  float* sk = (float*)d_ws;
  float* si = sk + n_nodes;

  // Phase 1: node projections
  const int tiles1 = n_nodes / 16;
  if (tiles1 > 0) {
    int blocks1 = (tiles1 + 7) / 8;   // 8 waves per 256-thread block
    node_proj_wmma<<<blocks1, 256, 0, stream>>>(h, W, sk, si, tiles1);
  }
  const int rem1 = n_nodes - tiles1 * 16;
  if (rem1 > 0)
    node_proj_tail<<<1, 32, 0, stream>>>(h, W, sk, si, tiles1 * 16, n_nodes);

  // Phase 2: edge outputs
  const int tiles2 = n_edges / 16;
  if (tiles2 > 0) {
    int blocks2 = (tiles2 + 3) / 4;   // 4 waves per 128-thread block
    edge_out_wmma<<<blocks2, 128, 0, stream>>>(e, W, b, src, dst, sk, si, p, tiles2);
  }
  const int rem2 = n_edges - tiles2 * 16;
  if (rem2 > 0)
    edge_out_tail<<<1, 32, 0, stream>>>(e, W, b, src, dst, sk, si, p,
                                        tiles2 * 16, n_edges);
}